// GNN_42786464203233
// MI455X (gfx1250) — compile-verified
//
#include <hip/hip_runtime.h>
#include <math.h>

typedef __attribute__((ext_vector_type(2))) float v2f;
typedef __attribute__((ext_vector_type(8))) float v8f;

// ---------------------------------------------------------------------------
// WMMA GEMM: C[M,N] = act(A[M,K] @ W[N,K]^T + bias), fp32 via V_WMMA_F32_16X16X4_F32
// One wave -> one 16x16 C tile. block = 256 (8 waves), grid = (ceil(N/16), ceil(M/128)).
// M must be a multiple of 16 (true here: 50000). K,N arbitrary.
// Steady-state loop is mask-free: a lane's accumulator holds exactly one output
// column, so out-of-range columns just compute garbage in lanes we never store
// (B row pointer clamped in-bounds). Only the K%4 tail needs (A-side) masking.
// EXEC is all-ones at every WMMA (per-wave-uniform early exit only).
// ---------------------------------------------------------------------------
template<bool RELU>
__global__ void wmma_gemm_bias(const float* __restrict__ A, const float* __restrict__ W,
                               const float* __restrict__ bias, float* __restrict__ C,
                               int M, int K, int N)
{
    const int lane = threadIdx.x & 31;
    const int wave = threadIdx.x >> 5;
    const int half = lane >> 4;          // 0: K slots {0,1}; 1: K slots {2,3}
    const int lm   = lane & 15;
    const int m0   = (blockIdx.y * 8 + wave) * 16;
    if (m0 >= M) return;                 // uniform per wave
    const int n0   = blockIdx.x * 16;
    const int col  = n0 + lm;
    const int colc = col < N ? col : N - 1;             // clamp only; no masking
    const float* __restrict__ arow = A + (size_t)(m0 + lm) * K + half * 2;
    const float* __restrict__ wrow = W + (size_t)colc * K + half * 2;

    v8f c = {};
    int k0 = 0;
    for (; k0 + 4 <= K; k0 += 4) {       // mask-free steady state
        v2f a, b;
        a.x = arow[k0];
        a.y = arow[k0 + 1];
        b.x = wrow[k0];
        b.y = wrow[k0 + 1];
        c = __builtin_amdgcn_wmma_f32_16x16x4_f32(false, a, false, b, (short)0, c,
                                                  false, false);
    }
    if (k0 < K) {                        // K%4 tail: zero A beyond K, clamp B addr
        const int ka = k0 + half * 2;
        const int kb = ka + 1;
        v2f a, b;
        a.x = (ka < K) ? A[(size_t)(m0 + lm) * K + ka] : 0.0f;
        a.y = (kb < K) ? A[(size_t)(m0 + lm) * K + kb] : 0.0f;
        b.x = W[(size_t)colc * K + (ka < K ? ka : 0)];
        b.y = W[(size_t)colc * K + (kb < K ? kb : 0)];
        c = __builtin_amdgcn_wmma_f32_16x16x4_f32(false, a, false, b, (short)0, c,
                                                  false, false);
    }
    if (col < N) {
        const float bv = bias[col];
        #pragma unroll
        for (int r = 0; r < 8; ++r) {
            float v = c[r] + bv;
            if (RELU) v = fmaxf(v, 0.0f);
            C[(size_t)(m0 + r + 8 * half) * N + col] = v;   // C row = r (+8 for hi half)
        }
    }
}

// ---------------------------------------------------------------------------
// Edge scatter: acc[dst] += x[src] (raw features), cnt[dst] += 1
// ---------------------------------------------------------------------------
template<int F>
__global__ void mp_scatter(const float* __restrict__ x, const int* __restrict__ src,
                           const int* __restrict__ dst, int E,
                           float* __restrict__ acc, float* __restrict__ cnt)
{
    long long t = (long long)blockIdx.x * blockDim.x + threadIdx.x;
    if (t >= (long long)E * F) return;
    int e = (int)(t / F);
    int f = (int)(t % F);
    int s = src[e], d = dst[e];
    atomicAdd(&acc[(size_t)d * F + f], x[(size_t)s * F + f]);
    if (f == 0) atomicAdd(&cnt[d], 1.0f);
}

// ---------------------------------------------------------------------------
// mp update: out = cnt>0 ? Wu*(Wn*(acc/cnt) + bn) + bu : bu
// (equals reference: mean of (Wn x_j + bn) then Wu*mean+bu; cnt==0 -> mean=0 -> bu)
// params layout: Wn[H,FIN], bn[H], Wu[FOUT,H], bu[FOUT]
// ---------------------------------------------------------------------------
template<int FIN, int H, int FOUT>
__global__ void mp_update(const float* __restrict__ acc, const float* __restrict__ cnt,
                          const float* __restrict__ params,
                          float* __restrict__ out, int n)
{
    __shared__ float sWn[H * FIN];
    __shared__ float sbn[H];
    __shared__ float sWu[FOUT * H];
    __shared__ float sbu[FOUT];
    const float* Wn = params;
    const float* bn = params + H * FIN;
    const float* Wu = bn + H;
    const float* bu = Wu + FOUT * H;
    for (int i = threadIdx.x; i < H * FIN;  i += blockDim.x) sWn[i] = Wn[i];
    for (int i = threadIdx.x; i < H;        i += blockDim.x) sbn[i] = bn[i];
    for (int i = threadIdx.x; i < FOUT * H; i += blockDim.x) sWu[i] = Wu[i];
    for (int i = threadIdx.x; i < FOUT;     i += blockDim.x) sbu[i] = bu[i];
    __syncthreads();
    int i = blockIdx.x * blockDim.x + threadIdx.x;
    if (i >= n) return;
    float cv = cnt[i];
    float o[FOUT];
    #pragma unroll
    for (int j = 0; j < FOUT; ++j) o[j] = sbu[j];
    if (cv > 0.0f) {
        const float inv = 1.0f / cv;     // cnt >= 1 here, so max(cnt,1)==cnt
        float mx[FIN];
        #pragma unroll
        for (int f = 0; f < FIN; ++f) mx[f] = acc[(size_t)i * FIN + f] * inv;
        for (int k = 0; k < H; ++k) {
            float h = sbn[k];
            #pragma unroll
            for (int f = 0; f < FIN; ++f) h += sWn[k * FIN + f] * mx[f];
            #pragma unroll
            for (int j = 0; j < FOUT; ++j) o[j] += sWu[j * H + k] * h;
        }
    }
    #pragma unroll
    for (int j = 0; j < FOUT; ++j) out[(size_t)i * FOUT + j] = o[j];
}

// ---------------------------------------------------------------------------
// Per-node 2-layer MLP (relu in middle), optional residual added to input.
// params: W1[H,FIN], b1[H], W2[FOUT,H], b2[FOUT]
// ---------------------------------------------------------------------------
template<int FIN, int H, int FOUT, bool RES>
__global__ void mlp_node(const float* __restrict__ x, const float* __restrict__ res,
                         const float* __restrict__ params,
                         float* __restrict__ out, int n)
{
    __shared__ float sW1[H * FIN];
    __shared__ float sb1[H];
    __shared__ float sW2[FOUT * H];
    __shared__ float sb2[FOUT];
    const float* W1 = params;
    const float* b1 = params + H * FIN;
    const float* W2 = b1 + H;
    const float* b2 = W2 + FOUT * H;
    for (int i = threadIdx.x; i < H * FIN;  i += blockDim.x) sW1[i] = W1[i];
    for (int i = threadIdx.x; i < H;        i += blockDim.x) sb1[i] = b1[i];
    for (int i = threadIdx.x; i < FOUT * H; i += blockDim.x) sW2[i] = W2[i];
    for (int i = threadIdx.x; i < FOUT;     i += blockDim.x) sb2[i] = b2[i];
    __syncthreads();
    int i = blockIdx.x * blockDim.x + threadIdx.x;
    if (i >= n) return;
    float in[FIN];
    #pragma unroll
    for (int f = 0; f < FIN; ++f) {
        float v = x[(size_t)i * FIN + f];
        if (RES) v += res[(size_t)i * FIN + f];
        in[f] = v;
    }
    float o[FOUT];
    #pragma unroll
    for (int j = 0; j < FOUT; ++j) o[j] = sb2[j];
    for (int k = 0; k < H; ++k) {
        float h = sb1[k];
        #pragma unroll
        for (int f = 0; f < FIN; ++f) h += sW1[k * FIN + f] * in[f];
        h = fmaxf(h, 0.0f);
        #pragma unroll
        for (int j = 0; j < FOUT; ++j) o[j] += sW2[j * H + k] * h;
    }
    #pragma unroll
    for (int j = 0; j < FOUT; ++j) out[(size_t)i * FOUT + j] = o[j];
}

// ---------------------------------------------------------------------------
// softmax over edge scores
// ---------------------------------------------------------------------------
__device__ __forceinline__ void atomicMaxF(float* addr, float v)
{
    unsigned int* ua = (unsigned int*)addr;
    unsigned int old = *ua;
    while (__uint_as_float(old) < v) {
        unsigned int assumed = old;
        old = atomicCAS(ua, assumed, __float_as_uint(v));
        if (old == assumed) break;
    }
}

__global__ void init_scalars_kernel(float* scal)
{
    if (threadIdx.x == 0 && blockIdx.x == 0) { scal[0] = -INFINITY; scal[1] = 0.0f; }
}

__global__ void scores_max_kernel(const float* __restrict__ d1, const float* __restrict__ d2,
                                  const int* __restrict__ ps, const int* __restrict__ pd,
                                  float* __restrict__ sc, int E, float* __restrict__ scal)
{
    __shared__ float red[256];
    int e = blockIdx.x * blockDim.x + threadIdx.x;
    float s = -INFINITY;
    if (e < E) {
        const float* a = d1 + (size_t)ps[e] * 8;
        const float* b = d2 + (size_t)pd[e] * 8;
        float acc = 0.0f;
        #pragma unroll
        for (int j = 0; j < 8; ++j) acc += a[j] * b[j];
        sc[e] = acc;
        s = acc;
    }
    red[threadIdx.x] = s;
    __syncthreads();
    for (int st = 128; st > 0; st >>= 1) {
        if (threadIdx.x < st) red[threadIdx.x] = fmaxf(red[threadIdx.x], red[threadIdx.x + st]);
        __syncthreads();
    }
    if (threadIdx.x == 0) atomicMaxF(&scal[0], red[0]);
}

__global__ void exp_sum_kernel(float* __restrict__ sc, float* __restrict__ scal, int E)
{
    __shared__ float red[256];
    int e = blockIdx.x * blockDim.x + threadIdx.x;
    const float gm = scal[0];
    float v = 0.0f;
    if (e < E) { v = expf(sc[e] - gm); sc[e] = v; }
    red[threadIdx.x] = v;
    __syncthreads();
    for (int st = 128; st > 0; st >>= 1) {
        if (threadIdx.x < st) red[threadIdx.x] += red[threadIdx.x + st];
        __syncthreads();
    }
    if (threadIdx.x == 0) atomicAdd(&scal[1], red[0]);
}

__global__ void normalize_kernel(float* __restrict__ sc, const float* __restrict__ scal, int E)
{
    int e = blockIdx.x * blockDim.x + threadIdx.x;
    if (e < E) sc[e] *= (1.0f / scal[1]);
}

// ---------------------------------------------------------------------------
extern "C" void kernel_launch(void* const* d_in, const int* in_sizes, int n_in,
                              void* d_out, int out_size, void* d_ws, size_t ws_size,
                              hipStream_t stream)
{
    (void)n_in; (void)ws_size;
    // dict order: 0 x_thief, 1 x_slot, 2 x_heist, 3 ei_conflicts, 4 ei_on, 5 ei_rev_on,
    // 6 ra_src, 7 ra_dst, 8 rp_src, 9 rp_dst, 10 p_src, 11 p_dst,
    // 12 enc1, 13 enc2, 14 enc3, 15 msg_a, 16 msg_b, 17 msg_c, 18 msg_d,
    // 19 msg_rev_a, 20 msg_rev_c, 21 dec1, 22 dec2
    const float* x_thief   = (const float*)d_in[0];
    const float* x_slot    = (const float*)d_in[1];
    const int*   ei_on     = (const int*)d_in[4];
    const int*   ei_rev_on = (const int*)d_in[5];
    const int*   rp_src    = (const int*)d_in[8];
    const int*   rp_dst    = (const int*)d_in[9];
    const int*   p_src     = (const int*)d_in[10];
    const int*   p_dst     = (const int*)d_in[11];
    const float* enc1      = (const float*)d_in[12];
    const float* enc2      = (const float*)d_in[13];
    const float* msg_a     = (const float*)d_in[15];
    const float* msg_c     = (const float*)d_in[17];
    const float* msg_rev_a = (const float*)d_in[19];
    const float* msg_rev_c = (const float*)d_in[20];
    const float* dec1      = (const float*)d_in[21];
    const float* dec2      = (const float*)d_in[22];

    const int N1 = in_sizes[0] / 111;          // 50000
    const int N2 = in_sizes[1] / 12;           // 100000
    const int N3 = in_sizes[2] / 4;            // 100000
    const int E_on  = in_sizes[4] / 2;         // 1,000,000
    const int E_pos = in_sizes[8];             // 1,000,000
    const int E_out = out_size;                // 1,000,000

    float* ws = (float*)d_ws;
    size_t off = 0;
    auto alloc = [&](size_t nf) { float* p = ws + off; off += nf; return p; };
    float* H1  = alloc((size_t)N1 * 132);      // enc1 hidden; reused as acc after enc1
    float* n1e = alloc((size_t)N1 * 64);
    float* n2e = alloc((size_t)N2 * 16);
    float* n3  = alloc((size_t)N3 * 8);
    float* n2m = alloc((size_t)N2 * 16);
    float* n1m = alloc((size_t)N1 * 64);
    float* n2r = alloc((size_t)N2 * 16);
    float* cnt = alloc((size_t)N2);            // max node count
    float* d1  = alloc((size_t)N1 * 8);
    float* d2  = alloc((size_t)N2 * 8);
    float* scal = alloc(8);
    float* acc = H1;                           // >= N2*64 floats, free after gemm2
    float* outp = (float*)d_out;

    const dim3 blk256(256);
    auto nb = [](long long n, int b) { return (unsigned)((n + b - 1) / b); };

    // ---- enc1 (WMMA fp32): X[50000,111] -> relu 132 -> 64 ----
    {
        const float* W1 = enc1;                 // [132,111]
        const float* b1 = enc1 + 132 * 111;
        const float* W2 = b1 + 132;             // [64,132]
        const float* b2 = W2 + 64 * 132;
        wmma_gemm_bias<true ><<<dim3(nb(132, 16), nb(N1, 128)), blk256, 0, stream>>>(
            x_thief, W1, b1, H1, N1, 111, 132);
        wmma_gemm_bias<false><<<dim3(nb(64, 16),  nb(N1, 128)), blk256, 0, stream>>>(
            H1, W2, b2, n1e, N1, 132, 64);
    }
    // ---- enc2: x_slot[100000,12] -> relu 32 -> 16 ----
    mlp_node<12, 32, 16, false><<<nb(N2, 256), blk256, 0, stream>>>(
        x_slot, nullptr, enc2, n2e, N2);

    // ---- mp stage 1: msg_rev_c, slot(n2e) -> heist, edges ei_on ----
    hipMemsetAsync(acc, 0, (size_t)N3 * 16 * sizeof(float), stream);
    hipMemsetAsync(cnt, 0, (size_t)N3 * sizeof(float), stream);
    mp_scatter<16><<<nb((long long)E_on * 16, 256), blk256, 0, stream>>>(
        n2e, ei_on, ei_on + E_on, E_on, acc, cnt);
    mp_update<16, 32, 8><<<nb(N3, 256), blk256, 0, stream>>>(acc, cnt, msg_rev_c, n3, N3);

    // ---- mp stage 2: msg_c, heist(n3) -> slot, edges ei_rev_on ----
    hipMemsetAsync(acc, 0, (size_t)N2 * 8 * sizeof(float), stream);
    hipMemsetAsync(cnt, 0, (size_t)N2 * sizeof(float), stream);
    mp_scatter<8><<<nb((long long)E_on * 8, 256), blk256, 0, stream>>>(
        n3, ei_rev_on, ei_rev_on + E_on, E_on, acc, cnt);
    mp_update<8, 16, 16><<<nb(N2, 256), blk256, 0, stream>>>(acc, cnt, msg_c, n2m, N2);

    // ---- mp stage 3: msg_a, slot(n2m) -> thief, edges rp ----
    hipMemsetAsync(acc, 0, (size_t)N1 * 16 * sizeof(float), stream);
    hipMemsetAsync(cnt, 0, (size_t)N1 * sizeof(float), stream);
    mp_scatter<16><<<nb((long long)E_pos * 16, 256), blk256, 0, stream>>>(
        n2m, rp_src, rp_dst, E_pos, acc, cnt);
    mp_update<16, 32, 64><<<nb(N1, 256), blk256, 0, stream>>>(acc, cnt, msg_a, n1m, N1);

    // ---- mp stage 4: msg_rev_a, thief(n1m) -> slot, edges p ----
    hipMemsetAsync(acc, 0, (size_t)N2 * 64 * sizeof(float), stream);
    hipMemsetAsync(cnt, 0, (size_t)N2 * sizeof(float), stream);
    mp_scatter<64><<<nb((long long)E_pos * 64, 256), blk256, 0, stream>>>(
        n1m, p_src, p_dst, E_pos, acc, cnt);
    mp_update<64, 128, 16><<<nb(N2, 256), blk256, 0, stream>>>(acc, cnt, msg_rev_a, n2r, N2);

    // ---- decoders with residual ----
    mlp_node<64, 32, 8, true><<<nb(N1, 256), blk256, 0, stream>>>(n1m, n1e, dec1, d1, N1);
    mlp_node<16, 12, 8, true><<<nb(N2, 256), blk256, 0, stream>>>(n2r, n2e, dec2, d2, N2);

    // ---- edge scores + softmax (3 passes, in-place in d_out) ----
    init_scalars_kernel<<<1, 1, 0, stream>>>(scal);
    scores_max_kernel<<<nb(E_out, 256), blk256, 0, stream>>>(d1, d2, p_src, p_dst,
                                                             outp, E_out, scal);
    exp_sum_kernel<<<nb(E_out, 256), blk256, 0, stream>>>(outp, scal, E_out);
    normalize_kernel<<<nb(E_out, 256), blk256, 0, stream>>>(outp, scal, E_out);
}